// SpikingBasalGanglia_13022340841688
// MI455X (gfx1250) — compile-verified
//
#include <hip/hip_runtime.h>

// ---------------------------------------------------------------------------
// Spiking basal ganglia scan, CDNA5 (gfx1250), wave32.
//
//   T=512 sequential LIF steps, B=256 circuits, 128 D1 + 128 D2 neurons,
//   STN/GPI collapse to per-circuit scalars (identical neurons, no noise).
//
// Mapping: block = 512 thr (16 waves) handles one 16-column batch tile.
//   wave w in [0,8):  d1 neuron tile w   (16 neurons x 16 circuits)
//   wave w in [8,16): d2 neuron tile w-8
// Per step, per wave: one V_WMMA_F32_16X16X4_F32 computes
//   v_new = (0.2*W_tile) x (x_t * dopamine_mod) + (0.8*v + 0.02*noise)
// i.e. GEMM + leak + noise fused; membrane state v stays resident in the
// WMMA accumulator layout (8 VGPRs/lane) for the whole T-scan.
// Spike means: shfl_xor half-wave fold + ds_add_f32 into double-buffered LDS,
// one barrier per step; wave 0 runs scalar STN/GPI and stores the gate.
// Noise (the only large stream: 128 MB total) is software-pipelined one step
// ahead in registers and prefetched 8 steps ahead (global_prefetch_b8).
// ---------------------------------------------------------------------------

typedef float v2f __attribute__((ext_vector_type(2)));
typedef float v8f __attribute__((ext_vector_type(8)));

#define T_STEPS 512
#define B_CIRC  256
#define N_POP   128          // neurons per d1/d2 population

__global__ __launch_bounds__(512, 1)
void SpikingBG_kernel(const float* __restrict__ x,      // [T,B,2]
                      const float* __restrict__ dopa,   // [T,B]
                      const float* __restrict__ Wd1,    // [128,2]
                      const float* __restrict__ Wd2,    // [128,2]
                      const float* __restrict__ nd1,    // [T,B,128]
                      const float* __restrict__ nd2,    // [T,B,128]
                      float* __restrict__ out)          // [T,B]
{
    __shared__ float sAcc[2][2][16];   // [phase][pop][batch-col] spike sums

    const int tid  = threadIdx.x;
    const int lane = tid & 31;
    const int wave = tid >> 5;
    const int pop  = wave >> 3;        // 0 = d1, 1 = d2
    const int nt   = wave & 7;         // neuron tile within population
    const int col  = lane & 15;        // batch column within tile (N index)
    const int half = lane >> 4;        // 0: K=0,1 / rows 0-7 ; 1: K=2,3 / rows 8-15
    const int b    = blockIdx.x * 16 + col;   // global circuit index

    if (tid < 64) ((float*)sAcc)[tid] = 0.0f;

    // --- A matrix: (1-TAU)*W tile, resident for the whole scan -------------
    // 16x4 f32 A layout: lanes 0-15 hold row M=lane, K=0 (v0) / K=1 (v1);
    // lanes 16-31 hold K=2,3 -> zero (K padded from 2 to 4).
    const float* W = pop ? Wd2 : Wd1;
    const int row = nt * 16 + col;
    v2f A;
    A.x = half ? 0.0f : 0.2f * W[row * 2 + 0];
    A.y = half ? 0.0f : 0.2f * W[row * 2 + 1];

    // --- noise stream: 8 consecutive floats / lane, matches C/D layout -----
    // lane holds rows M = nt*16 + half*8 + r (r=0..7) of column b.
    const float* noise = pop ? nd2 : nd1;
    const float4* np4 =
        reinterpret_cast<const float4*>(noise + (size_t)b * N_POP + nt * 16 + half * 8);
    const size_t tstride4 = (size_t)B_CIRC * N_POP / 4;   // float4 per step

    // preload t = 0 (software pipeline depth 1)
    float4 n0 = np4[0];
    float4 n1 = np4[1];
    float  x0 = x[(size_t)b * 2 + 0];
    float  x1 = x[(size_t)b * 2 + 1];
    float  dp = dopa[b];

    v8f   v     = {};        // membrane state, WMMA accumulator layout
    float v_stn = 0.0f;      // wave-0 scalar populations (lanes 0-15)
    float v_gpi = 0.0f;

    __syncthreads();

    for (int t = 0; t < T_STEPS; ++t) {
        // stage far-ahead noise lines into cache (global_prefetch_b8)
        if (t + 8 < T_STEPS)
            __builtin_prefetch(reinterpret_cast<const char*>(np4 + (size_t)(t + 8) * tstride4), 0, 0);

        const float4 c0 = n0, c1 = n1;
        const float  cx0 = x0, cx1 = x1, cdp = dp;
        if (t + 1 < T_STEPS) {                       // pipeline next step
            n0 = np4[(size_t)(t + 1) * tstride4 + 0];
            n1 = np4[(size_t)(t + 1) * tstride4 + 1];
            const size_t xo = (size_t)(t + 1) * B_CIRC + b;
            x0 = x[xo * 2 + 0];
            x1 = x[xo * 2 + 1];
            dp = dopa[xo];
        }

        // --- B matrix: dopamine-modulated input column (K=2,3 -> zero) ----
        const float mod = pop ? (1.0f - 0.3f * cdp) : (1.0f + 0.5f * cdp);
        v2f Bv;
        Bv.x = half ? 0.0f : cx0 * mod;
        Bv.y = half ? 0.0f : cx1 * mod;

        // --- C = TAU*v + (1-TAU)*NOISE*noise  (leak + noise injection) ----
        v8f C;
        C[0] = 0.8f * v[0] + 0.02f * c0.x;
        C[1] = 0.8f * v[1] + 0.02f * c0.y;
        C[2] = 0.8f * v[2] + 0.02f * c0.z;
        C[3] = 0.8f * v[3] + 0.02f * c0.w;
        C[4] = 0.8f * v[4] + 0.02f * c1.x;
        C[5] = 0.8f * v[5] + 0.02f * c1.y;
        C[6] = 0.8f * v[6] + 0.02f * c1.z;
        C[7] = 0.8f * v[7] + 0.02f * c1.w;

        // --- fused GEMM + LIF integrate: one 16x16x4 f32 WMMA -------------
        v8f vn = __builtin_amdgcn_wmma_f32_16x16x4_f32(
            /*neg_a=*/false, A, /*neg_b=*/false, Bv,
            /*c_mod=*/(short)0, C, /*reuse_a=*/false, /*reuse_b=*/false);

        // --- threshold / reset / local spike count ------------------------
        float localSum = 0.0f;
#pragma unroll
        for (int r = 0; r < 8; ++r) {
            const float s = (vn[r] >= 0.5f) ? 1.0f : 0.0f;
            localSum += s;
            v[r] = vn[r] * (1.0f - s);
        }

        // fold the two half-wave rows of each column, then one ds_add_f32
        const float pairSum = localSum + __shfl_xor(localSum, 16, 32);
        const int p = t & 1;
        if (half == 0) atomicAdd(&sAcc[p][pop][col], pairSum);

        __syncthreads();

        // --- wave 0: scalar STN/GPI dynamics + gate output ----------------
        if (wave == 0) {
            const float sum_d1 = sAcc[p][0][col];
            const float sum_d2 = sAcc[p][1][col];
            sAcc[p][0][col] = 0.0f;    // reset phase p for step t+2
            sAcc[p][1][col] = 0.0f;
            const float m1 = sum_d1 * (1.0f / 128.0f);
            const float m2 = sum_d2 * (1.0f / 128.0f);
            // STN (64 identical neurons -> scalar): I = mean(s_d2)*0.5
            float vs = 0.8f * v_stn + 0.2f * (m2 * 0.5f);
            const float ss = (vs >= 0.5f) ? 1.0f : 0.0f;
            v_stn = vs * (1.0f - ss);
            // GPI: I = 0.4 - 0.8*mean(s_d1) + 0.6*s_stn
            float vg = 0.8f * v_gpi + 0.2f * (0.4f - 0.8f * m1 + 0.6f * ss);
            const float sg = (vg >= 0.5f) ? 1.0f : 0.0f;
            v_gpi = vg * (1.0f - sg);
            if (half == 0) out[(size_t)t * B_CIRC + b] = m1 - sg;
        }
        // no second barrier needed: other waves next touch sAcc[p] only at
        // t+2, after the t+1 barrier, which wave 0 signals only after the
        // reset above has completed (program order).
    }
}

extern "C" void kernel_launch(void* const* d_in, const int* in_sizes, int n_in,
                              void* d_out, int out_size, void* d_ws, size_t ws_size,
                              hipStream_t stream) {
    const float* x    = (const float*)d_in[0];   // [512,256,2]
    const float* dopa = (const float*)d_in[1];   // [512,256]
    // d_in[2] = rpe — unused by the reference
    const float* Wd1  = (const float*)d_in[3];   // [128,2]
    const float* Wd2  = (const float*)d_in[4];   // [128,2]
    const float* nd1  = (const float*)d_in[5];   // [512,256,128]
    const float* nd2  = (const float*)d_in[6];   // [512,256,128]
    float* out = (float*)d_out;                  // [512,256]

    SpikingBG_kernel<<<B_CIRC / 16, 512, 0, stream>>>(x, dopa, Wd1, Wd2, nd1, nd2, out);
}